// Material_Attention_69544110456917
// MI455X (gfx1250) — compile-verified
//
#include <hip/hip_runtime.h>
#include <hip/hip_bf16.h>
#include <math.h>

// Problem dims (fixed by the reference)
#define BATCH 16
#define CH    256
#define CL    32
#define HW    16384          // 128*128
#define EPSV  1e-5f

typedef __attribute__((ext_vector_type(2))) float v2f;
typedef __attribute__((ext_vector_type(8))) float v8f;

// ---------------------------------------------------------------------------
// Pass 1: per-plane mean+max pooling.  grid = B*C planes, 256 threads/plane.
// Each thread streams 16 x b128 loads (64 floats) -> wave32 shuffle reduce ->
// LDS cross-wave reduce.  Bandwidth-bound: reads the full 256 MB of feat once.
// ---------------------------------------------------------------------------
__global__ void __launch_bounds__(256)
pool_kernel(const float* __restrict__ feat, float* __restrict__ pooled) {
    const int plane = blockIdx.x;                       // b*C + c
    const float4* src = (const float4*)(feat + (size_t)plane * HW);
    float s = 0.0f;
    float m = -__builtin_inff();
#pragma unroll
    for (int i = 0; i < 16; ++i) {
        float4 v = src[threadIdx.x + i * 256];
        s += (v.x + v.y) + (v.z + v.w);
        m = fmaxf(m, fmaxf(fmaxf(v.x, v.y), fmaxf(v.z, v.w)));
    }
    // wave32 reduction
#pragma unroll
    for (int off = 16; off > 0; off >>= 1) {
        s += __shfl_down(s, off, 32);
        m = fmaxf(m, __shfl_down(m, off, 32));
    }
    __shared__ float ls[8], lm[8];
    const int wave = threadIdx.x >> 5, lane = threadIdx.x & 31;
    if (lane == 0) { ls[wave] = s; lm[wave] = m; }
    __syncthreads();
    if (threadIdx.x == 0) {
        float S = ls[0], M = lm[0];
#pragma unroll
        for (int w = 1; w < 8; ++w) { S += ls[w]; M = fmaxf(M, lm[w]); }
        pooled[plane] = S * (1.0f / (float)HW) + M;     // mean + max
    }
}

// ---------------------------------------------------------------------------
// Pass 2: the whole attention head in one 2-wave block using
// V_WMMA_F32_16X16X4_F32 (fp32 in / fp32 out -> no precision loss).
//   matmul1: pooled[16,256] @ w_down^T[256,32]  (wave w owns N-tile 16*w)
//   softmax self-attention over Cl=32 (plain VALU, trivial)
//   matmul2: node[16,32] @ w_up^T[32,256]       (wave w owns 8 N-tiles)
// Fragment layouts per ISA 7.12.2:
//   A (16x4 f32): vgpr0 = K=k   (lanes 0-15) / K=k+2 (lanes 16-31), M = lane%16
//                 vgpr1 = K=k+1 / K=k+3
//   B (4x16 f32): vgpr0 = row K=k (lanes 0-15 = N) / row K=k+2 (lanes 16-31)
//                 vgpr1 = row K=k+1 / K=k+3
//   C/D (16x16):  vgpr r, lanes 0-15 -> M=r, lanes 16-31 -> M=r+8, N = lane%16
// ---------------------------------------------------------------------------
__global__ void __launch_bounds__(64)
attn_kernel(const float* __restrict__ pooled,
            const float* __restrict__ w_down,
            const float* __restrict__ g1, const float* __restrict__ b1,
            const float* __restrict__ rm1, const float* __restrict__ rv1,
            const float* __restrict__ w_up,
            const float* __restrict__ g2, const float* __restrict__ b2,
            const float* __restrict__ rm2, const float* __restrict__ rv2,
            float* __restrict__ att) {
    __shared__ float sp[BATCH * CH];     // pooled, 16 KB
    __shared__ float slat[BATCH * CL];   // latent
    __shared__ float snode[BATCH * CL];  // node

    const int tid  = threadIdx.x;        // 0..63
    const int wave = tid >> 5;           // 0..1
    const int lane = tid & 31;
    const int half = lane >> 4;          // lane group (0: lanes 0-15, 1: 16-31)
    const int l    = lane & 15;

    for (int i = tid; i < BATCH * CH; i += 64) sp[i] = pooled[i];
    __syncthreads();

    // ---- matmul1 + BN1 + ReLU -> slat[16][32] ----
    {
        const int nb = wave * 16;                       // N tile base (Cl dim)
        v8f acc = {};
        for (int k = 0; k < CH; k += 4) {
            v2f a, b;
            a.x = sp[l * CH + k + 2 * half];            // A[M=l][K]
            a.y = sp[l * CH + k + 2 * half + 1];
            // B[kk][n] = w_down[n][kk]; w_down is [CL][CH] row-major
            b.x = w_down[(nb + l) * CH + k + 2 * half];
            b.y = w_down[(nb + l) * CH + k + 2 * half + 1];
            acc = __builtin_amdgcn_wmma_f32_16x16x4_f32(
                false, a, false, b, (short)0, acc, false, false);
        }
        const int cn = nb + l;                          // Cl channel
        const float sc = g1[cn] / sqrtf(rv1[cn] + EPSV);
        const float mu = rm1[cn], bof = b1[cn];
#pragma unroll
        for (int r = 0; r < 8; ++r) {
            float v = (acc[r] - mu) * sc + bof;
            slat[(r + 8 * half) * CL + cn] = fmaxf(v, 0.0f);
        }
    }
    __syncthreads();

    // ---- per-batch 32-wide softmax self-attention -> snode[16][32] ----
    for (int p = tid; p < BATCH * CL; p += 64) {
        const int mrow = p >> 5, i = p & 31;
        const float* lr = &slat[mrow * CL];
        const float li = lr[i];
        float mx = -__builtin_inff();
#pragma unroll
        for (int j = 0; j < CL; ++j) mx = fmaxf(mx, li * lr[j]);
        float se = 0.0f, sw = 0.0f;
#pragma unroll
        for (int j = 0; j < CL; ++j) {
            const float e = __expf(li * lr[j] - mx);
            se += e; sw += e * lr[j];
        }
        snode[p] = sw / se;
    }
    __syncthreads();

    // ---- matmul2 + BN2 + ReLU + sigmoid -> att[16][256] (workspace) ----
    for (int t = wave; t < 16; t += 2) {                // uniform per wave
        const int nb = t * 16;                          // N tile base (C dim)
        v8f acc = {};
#pragma unroll
        for (int k = 0; k < CL; k += 4) {
            v2f a, b;
            a.x = snode[l * CL + k + 2 * half];
            a.y = snode[l * CL + k + 2 * half + 1];
            // B[kk][n] = w_up[n][kk]; w_up is [CH][CL] row-major
            b.x = w_up[(nb + l) * CL + k + 2 * half];
            b.y = w_up[(nb + l) * CL + k + 2 * half + 1];
            acc = __builtin_amdgcn_wmma_f32_16x16x4_f32(
                false, a, false, b, (short)0, acc, false, false);
        }
        const int cn = nb + l;                          // C channel
        const float sc = g2[cn] / sqrtf(rv2[cn] + EPSV);
        const float mu = rm2[cn], bof = b2[cn];
#pragma unroll
        for (int r = 0; r < 8; ++r) {
            float v = (acc[r] - mu) * sc + bof;
            v = fmaxf(v, 0.0f);
            v = 1.0f / (1.0f + __expf(-v));             // sigmoid
            att[(r + 8 * half) * CH + cn] = v;
        }
    }
}

// ---------------------------------------------------------------------------
// Pass 3: out = feat * (1 + att[b,c]).  Same streaming shape as pass 1.
// ---------------------------------------------------------------------------
__global__ void __launch_bounds__(256)
scale_kernel(const float* __restrict__ feat, const float* __restrict__ att,
             float* __restrict__ out) {
    const int plane = blockIdx.x;                       // b*C + c
    const float s = 1.0f + att[plane];                  // uniform -> scalarized
    const float4* src = (const float4*)(feat + (size_t)plane * HW);
    float4* dst       = (float4*)(out  + (size_t)plane * HW);
#pragma unroll
    for (int i = 0; i < 16; ++i) {
        float4 v = src[threadIdx.x + i * 256];
        v.x *= s; v.y *= s; v.z *= s; v.w *= s;
        dst[threadIdx.x + i * 256] = v;
    }
}

extern "C" void kernel_launch(void* const* d_in, const int* in_sizes, int n_in,
                              void* d_out, int out_size, void* d_ws, size_t ws_size,
                              hipStream_t stream) {
    const float* feat   = (const float*)d_in[0];
    const float* w_down = (const float*)d_in[1];
    const float* g1     = (const float*)d_in[2];
    const float* b1     = (const float*)d_in[3];
    const float* rm1    = (const float*)d_in[4];
    const float* rv1    = (const float*)d_in[5];
    const float* w_up   = (const float*)d_in[6];
    const float* g2     = (const float*)d_in[7];
    const float* b2     = (const float*)d_in[8];
    const float* rm2    = (const float*)d_in[9];
    const float* rv2    = (const float*)d_in[10];
    float* out = (float*)d_out;

    float* pooled = (float*)d_ws;                 // [16*256]
    float* att    = pooled + BATCH * CH;          // [16*256]

    pool_kernel <<<BATCH * CH, 256, 0, stream>>>(feat, pooled);
    attn_kernel <<<1, 64, 0, stream>>>(pooled, w_down, g1, b1, rm1, rv1,
                                       w_up, g2, b2, rm2, rv2, att);
    scale_kernel<<<BATCH * CH, 256, 0, stream>>>(feat, att, out);
}